// MemSE_54125177864243
// MI455X (gfx1250) — compile-verified
//
#include <hip/hip_runtime.h>
#include <hip/hip_bf16.h>
#include <math.h>

typedef __attribute__((ext_vector_type(2))) float v2f;
typedef __attribute__((ext_vector_type(8))) float v8f;

#define BATCH 4
#define CCH   3
#define HH    64
#define WW    64
#define D1    3072
#define D2    2048
#define D3    1024
#define SIGMA 0.01f

#define BK     32            // K-block staged through LDS
#define NKB    (D2 / BK)     // 64 K-blocks
#define LDS_S  132           // K-row stride in floats (128 + 4 pad; 2-way max conflicts)

// ---------------- Kernel 1: bias + 2x2 avg-pool -> xp[B, D1] ----------------
__global__ void pool_kernel(const float* __restrict__ x, const float* __restrict__ bias,
                            float* __restrict__ xp) {
    int t = blockIdx.x * 256 + threadIdx.x;
    if (t >= BATCH * D1) return;
    int b = t / D1, f = t % D1;
    int c = f >> 10;
    int rem = f & 1023;
    int oh = rem >> 5, ow = rem & 31;
    int base = c * (HH * WW) + (2 * oh) * WW + 2 * ow;
    const float* xb = x + b * (CCH * HH * WW);
    float s = (xb[base]        + bias[base])
            + (xb[base + 1]    + bias[base + 1])
            + (xb[base + WW]   + bias[base + WW])
            + (xb[base + WW+1] + bias[base + WW + 1]);
    xp[t] = 0.25f * s;
}

// ---------------- Kernel 2/3: column sum of |W| -----------------------------
__global__ void colabs_kernel(const float* __restrict__ W, float* __restrict__ out,
                              int rows, int cols) {
    int i = blockIdx.x * 256 + threadIdx.x;
    if (i >= cols) return;
    float s = 0.f;
    for (int o = 0; o < rows; ++o) s += fabsf(W[(size_t)o * cols + i]);
    out[i] = s;
}

// -------- GEMV: out[b, o] = sum_i W[o,i] * X[b,i], one block per o ----------
__global__ void gemv4_kernel(const float* __restrict__ W, const float* __restrict__ X,
                             float* __restrict__ out, int din, int dout) {
    __shared__ float red[4][256];
    int o = blockIdx.x, tid = threadIdx.x;
    const float* wr = W + (size_t)o * din;
    float a0 = 0.f, a1 = 0.f, a2 = 0.f, a3 = 0.f;
    for (int i = tid; i < din; i += 256) {
        float w = wr[i];
        a0 += w * X[i];
        a1 += w * X[din + i];
        a2 += w * X[2 * din + i];
        a3 += w * X[3 * din + i];
    }
    red[0][tid] = a0; red[1][tid] = a1; red[2][tid] = a2; red[3][tid] = a3;
    __syncthreads();
    for (int s = 128; s > 0; s >>= 1) {
        if (tid < s) {
            red[0][tid] += red[0][tid + s];
            red[1][tid] += red[1][tid + s];
            red[2][tid] += red[2][tid + s];
            red[3][tid] += red[3][tid + s];
        }
        __syncthreads();
    }
    if (tid < 4) out[tid * dout + o] = red[tid][0];
}

// ------------- Kernel 5: vn1[b], P1[b] from xp^2 and colabs1 ----------------
__global__ void stats1_kernel(const float* __restrict__ xp, const float* __restrict__ colabs1,
                              const float* __restrict__ g1p, float* __restrict__ vn1,
                              float* __restrict__ P1) {
    __shared__ float s1[256], s2[256];
    int b = blockIdx.x, tid = threadIdx.x;
    float m = 0.f, p = 0.f;
    for (int i = tid; i < D1; i += 256) {
        float v = xp[b * D1 + i];
        float v2 = v * v;
        m += v2;
        p += colabs1[i] * v2;
    }
    s1[tid] = m; s2[tid] = p;
    __syncthreads();
    for (int s = 128; s > 0; s >>= 1) {
        if (tid < s) { s1[tid] += s1[tid + s]; s2[tid] += s2[tid + s]; }
        __syncthreads();
    }
    if (tid == 0) {
        float c1 = g1p[0];                          // c1 = g1 / WMAX1, WMAX1 = 1
        vn1[b] = 2.f * (SIGMA * SIGMA) / (c1 * c1) * s1[0];
        P1[b]  = c1 * s2[0];                        // R = 1
    }
}

// ------------- Kernel 6: d^2, softplus, m2b ---------------------------------
__global__ void act_kernel(const float* __restrict__ mu1, const float* __restrict__ vn1,
                           float* __restrict__ d2, float* __restrict__ sp,
                           float* __restrict__ m2b) {
    int t = blockIdx.x * 256 + threadIdx.x;
    if (t >= BATCH * D2) return;
    int b = t / D2;
    float m = mu1[t];
    float dd = 1.f / (1.f + expf(-m));              // sigmoid
    float s  = (m > 20.f) ? m : log1pf(expf(m));    // softplus
    float dsq = dd * dd;
    d2[t] = dsq;
    sp[t] = s;
    m2b[t] = s * s + vn1[b] * dsq;
}

// ------------- Kernel 7: vn2[b], P_tot[b] -> output tail --------------------
__global__ void stats2_kernel(const float* __restrict__ m2b, const float* __restrict__ colabs2,
                              const float* __restrict__ g2p, const float* __restrict__ P1,
                              float* __restrict__ vn2, float* __restrict__ Pout) {
    __shared__ float s1[256], s2[256];
    int b = blockIdx.x, tid = threadIdx.x;
    float m = 0.f, p = 0.f;
    for (int j = tid; j < D2; j += 256) {
        float v = m2b[b * D2 + j];
        m += v;
        p += colabs2[j] * v;
    }
    s1[tid] = m; s2[tid] = p;
    __syncthreads();
    for (int s = 128; s > 0; s >>= 1) {
        if (tid < s) { s1[tid] += s1[tid + s]; s2[tid] += s2[tid + s]; }
        __syncthreads();
    }
    if (tid == 0) {
        float c2 = g2p[0];
        vn2[b] = 2.f * (SIGMA * SIGMA) / (c2 * c2) * s1[0];
        Pout[b] = P1[b] + c2 * s2[0];
    }
}

// ---------------------------------------------------------------------------
// Kernel 9: gamma2[b] = vn1_b * (W2*diag(d2_b)) W2^T + vn2_b I
//
// 128-thread block (4 waves) -> 128x128 output tile; wave (wy,wx) owns a
// 64x64 sub-tile as a 4x4 grid of V_WMMA_F32_16X16X4_F32 accumulators.
// K is staged through LDS in KB=32 blocks, stored K-major (element (k,row) at
// k*LDS_S + row) so that:
//   - global loads are coalesced b128 along K (16 per thread per K-block),
//   - fragment reads are lane-consecutive ds_load_b32 (<=2-way conflicts),
//   - d^2 is folded into the A tile once at staging time.
// Next K-block is register-prefetched during the compute phase.
//
// f32 WMMA fragment layout (ISA 7.12.2): lane<16 holds K=k,k+1 ; lane>=16
// holds K=k+2,k+3; rows/cols = lane&15. A and B tiles both come from W2 rows.
// ---------------------------------------------------------------------------
__global__ void __launch_bounds__(128)
gamma_kernel(const float* __restrict__ W2, const float* __restrict__ d2,
             const float* __restrict__ vn1, const float* __restrict__ vn2,
             float* __restrict__ out) {
    __shared__ float As[BK * LDS_S];
    __shared__ float Bs[BK * LDS_S];

    const int tid  = threadIdx.x;
    const int lane = tid & 31;
    const int w    = tid >> 5;          // wave 0..3
    const int wy   = w >> 1;
    const int wx   = w & 1;
    const int r    = lane & 15;
    const int hi   = lane >> 4;

    const int brow = blockIdx.x * 128;
    const int bcol = blockIdx.y * 128;
    const int b    = blockIdx.z;

    // staging mapping: thread -> (row = p*16 + tid/8, float4 column c4)
    const int srow = tid >> 3;          // 0..15
    const int c4   = tid & 7;           // 0..7  (covers 32 K columns)
    const float* d2b = d2 + (size_t)b * D2;
    const float* gA  = W2 + (size_t)(brow + srow) * D2 + c4 * 4;
    const float* gB  = W2 + (size_t)(bcol + srow) * D2 + c4 * 4;

    float4 ra[8], rb[8], rd;

    // prologue: load K-block 0 into registers
    rd = *(const float4*)(d2b + c4 * 4);
#pragma unroll
    for (int p = 0; p < 8; ++p) {
        ra[p] = *(const float4*)(gA + (size_t)p * 16 * D2);
        rb[p] = *(const float4*)(gB + (size_t)p * 16 * D2);
    }

    v8f acc[4][4] = {};

    const int ca = wy * 64 + r;         // wave's A column base in LDS tile
    const int cb = wx * 64 + r;         // wave's B column base in LDS tile

    for (int kb = 0; kb < NKB; ++kb) {
        __syncthreads();                // previous compute done reading LDS
        // stage registers -> LDS (K-major), folding d^2 into A
#pragma unroll
        for (int p = 0; p < 8; ++p) {
            int sbase = c4 * 4 * LDS_S + p * 16 + srow;
            As[sbase            ] = ra[p].x * rd.x;
            As[sbase +     LDS_S] = ra[p].y * rd.y;
            As[sbase + 2 * LDS_S] = ra[p].z * rd.z;
            As[sbase + 3 * LDS_S] = ra[p].w * rd.w;
            Bs[sbase            ] = rb[p].x;
            Bs[sbase +     LDS_S] = rb[p].y;
            Bs[sbase + 2 * LDS_S] = rb[p].z;
            Bs[sbase + 3 * LDS_S] = rb[p].w;
        }
        // register-prefetch next K-block (overlaps with compute below)
        if (kb + 1 < NKB) {
            const float* gA2 = gA + (kb + 1) * BK;
            const float* gB2 = gB + (kb + 1) * BK;
            rd = *(const float4*)(d2b + (kb + 1) * BK + c4 * 4);
#pragma unroll
            for (int p = 0; p < 8; ++p) {
                ra[p] = *(const float4*)(gA2 + (size_t)p * 16 * D2);
                rb[p] = *(const float4*)(gB2 + (size_t)p * 16 * D2);
            }
        }
        __syncthreads();                // LDS tile ready

#pragma unroll
        for (int kk = 0; kk < BK; kk += 4) {
            const int k0 = (kk + 2 * hi) * LDS_S;
            v2f a[4], bv[4];
#pragma unroll
            for (int i = 0; i < 4; ++i) {
                a[i][0]  = As[k0         + ca + i * 16];
                a[i][1]  = As[k0 + LDS_S + ca + i * 16];
                bv[i][0] = Bs[k0         + cb + i * 16];
                bv[i][1] = Bs[k0 + LDS_S + cb + i * 16];
            }
#pragma unroll
            for (int i = 0; i < 4; ++i)
#pragma unroll
                for (int j = 0; j < 4; ++j)
                    acc[i][j] = __builtin_amdgcn_wmma_f32_16x16x4_f32(
                        false, a[i], false, bv[j], (short)0, acc[i][j], false, false);
        }
    }

    // epilogue: scale by vn1, add vn2 on diagonal, store
    const float s  = vn1[b];
    const float vd = vn2[b];
    float* ob = out + (size_t)b * D3 * D3;
    // C/D layout: (vgpr v, lane l): row = v + 8*(l>>4), col = l&15
#pragma unroll
    for (int i = 0; i < 4; ++i) {
#pragma unroll
        for (int j = 0; j < 4; ++j) {
#pragma unroll
            for (int v = 0; v < 8; ++v) {
                int row = brow + wy * 64 + i * 16 + v + 8 * hi;
                int col = bcol + wx * 64 + j * 16 + r;
                float val = s * acc[i][j][v];
                if (row == col) val += vd;
                ob[(size_t)row * D3 + col] = val;
            }
        }
    }
}

// ---------------------------------------------------------------------------
extern "C" void kernel_launch(void* const* d_in, const int* in_sizes, int n_in,
                              void* d_out, int out_size, void* d_ws, size_t ws_size,
                              hipStream_t stream) {
    (void)in_sizes; (void)n_in; (void)out_size; (void)ws_size;
    const float* x    = (const float*)d_in[0];   // [4,3,64,64]
    const float* bias = (const float*)d_in[1];   // [3,64,64]
    const float* W1   = (const float*)d_in[2];   // [2048,3072]
    const float* W2   = (const float*)d_in[3];   // [1024,2048]
    const float* g1   = (const float*)d_in[4];   // scalar
    const float* g2   = (const float*)d_in[5];   // scalar

    float* out = (float*)d_out;
    float* out_mu2   = out;                       // 4*1024
    float* out_gamma = out + BATCH * D3;          // 4*1024*1024
    float* out_P     = out + BATCH * D3 + (size_t)BATCH * D3 * D3;  // 4

    float* ws = (float*)d_ws;
    float* xp      = ws;                 // 4*3072
    float* colabs1 = xp + BATCH * D1;    // 3072
    float* colabs2 = colabs1 + D1;       // 2048
    float* mu1     = colabs2 + D2;       // 4*2048
    float* d2      = mu1 + BATCH * D2;   // 4*2048
    float* sp      = d2 + BATCH * D2;    // 4*2048
    float* m2b     = sp + BATCH * D2;    // 4*2048
    float* vn1     = m2b + BATCH * D2;   // 4
    float* vn2     = vn1 + BATCH;        // 4
    float* P1      = vn2 + BATCH;        // 4

    pool_kernel<<<(BATCH * D1 + 255) / 256, 256, 0, stream>>>(x, bias, xp);
    colabs_kernel<<<(D1 + 255) / 256, 256, 0, stream>>>(W1, colabs1, D2, D1);
    colabs_kernel<<<(D2 + 255) / 256, 256, 0, stream>>>(W2, colabs2, D3, D2);
    gemv4_kernel<<<D2, 256, 0, stream>>>(W1, xp, mu1, D1, D2);
    stats1_kernel<<<BATCH, 256, 0, stream>>>(xp, colabs1, g1, vn1, P1);
    act_kernel<<<(BATCH * D2 + 255) / 256, 256, 0, stream>>>(mu1, vn1, d2, sp, m2b);
    stats2_kernel<<<BATCH, 256, 0, stream>>>(m2b, colabs2, g2, P1, vn2, out_P);
    gemv4_kernel<<<D3, 256, 0, stream>>>(W2, sp, out_mu2, D2, D3);

    dim3 ggrid(D3 / 128, D3 / 128, BATCH);   // 8 x 8 x 4 tiles of 128x128
    gamma_kernel<<<ggrid, 128, 0, stream>>>(W2, d2, vn1, vn2, out_gamma);
}